// ResGAT_34574486733589
// MI455X (gfx1250) — compile-verified
//
#include <hip/hip_runtime.h>
#include <hip/hip_bf16.h>
#include <math.h>

// ---------------------------------------------------------------------------
// ResGAT forward for gfx1250 (MI455X).
//  * GEMMs run on V_WMMA_F32_16X16X4_F32 (full fp32 — GEMM is ~0.1% of the
//    runtime roofline, so precision > matrix throughput here).
//  * Edge softmax/scatter uses wave-per-edge coalesced float atomics.
//  * All feature-dim ops are laid out for wave32 (32-lane shuffles, 4-wave
//    128-thread column blocks).
// ---------------------------------------------------------------------------

typedef __attribute__((ext_vector_type(2))) float v2f;
typedef __attribute__((ext_vector_type(8))) float v8f;

#define HID 128
#define NHEAD 4
#define CH 32
#define EPSV 1e-5f

// ------------------------------- utilities ---------------------------------

__global__ void zero_f32_k(float* p, long n) {
  for (long i = blockIdx.x * (long)blockDim.x + threadIdx.x; i < n;
       i += (long)gridDim.x * blockDim.x)
    p[i] = 0.0f;
}

__device__ __forceinline__ float leaky02(float x) {
  return x > 0.0f ? x : 0.2f * x;
}

// order-preserving float->uint key for atomicMax-based segment_max
__device__ __forceinline__ unsigned fkey(float f) {
  int i = __float_as_int(f);
  return (unsigned)(i ^ ((i >> 31) | 0x80000000));
}
__device__ __forceinline__ float funkey(unsigned u) {
  int i = (int)((u & 0x80000000u) ? (u ^ 0x80000000u) : ~u);
  return __int_as_float(i);
}

// --------------------------- batchnorm (axis=0) ----------------------------

// thread t owns column t; rows swept coalesced; partials folded via atomics
__global__ __launch_bounds__(128) void bn_partial_k(const float* __restrict__ x,
                                                    int nrows,
                                                    float* __restrict__ stats) {
  const int c = threadIdx.x;  // 0..127
  float s = 0.0f, s2 = 0.0f;
  for (long r = blockIdx.x; r < nrows; r += gridDim.x) {
    float v = x[r * HID + c];
    s += v;
    s2 += v * v;
  }
  atomicAdd(&stats[c], s);
  atomicAdd(&stats[HID + c], s2);
}

__global__ __launch_bounds__(128) void bn_finalize_k(const float* __restrict__ stats,
                                                     const float* __restrict__ g,
                                                     const float* __restrict__ b,
                                                     float inv_n,
                                                     float* __restrict__ ss) {
  const int c = threadIdx.x;
  float mean = stats[c] * inv_n;
  float var = stats[HID + c] * inv_n - mean * mean;
  float sc = g[c] * rsqrtf(var + EPSV);
  ss[c] = sc;
  ss[HID + c] = b[c] - mean * sc;
}

__global__ void bn_apply_k(const float* __restrict__ x,
                           const float* __restrict__ ss, float* __restrict__ y,
                           long total) {
  for (long i = blockIdx.x * (long)blockDim.x + threadIdx.x; i < total;
       i += (long)gridDim.x * blockDim.x) {
    int c = (int)(i & (HID - 1));
    y[i] = x[i] * ss[c] + ss[HID + c];
  }
}

// ------------------------------- layernorm ---------------------------------

// one wave (32 lanes) per 128-wide row; float4 per lane; shfl_xor reduction
__global__ __launch_bounds__(256) void layernorm128_k(
    const float* __restrict__ x, const float* __restrict__ g,
    const float* __restrict__ b, float* __restrict__ y, int nrows, int relu) {
  const int lane = threadIdx.x & 31;
  const long row = (long)blockIdx.x * (blockDim.x >> 5) + (threadIdx.x >> 5);
  if (row >= nrows) return;
  const float4 v = ((const float4*)(x + row * HID))[lane];
  float s = v.x + v.y + v.z + v.w;
  float s2 = v.x * v.x + v.y * v.y + v.z * v.z + v.w * v.w;
#pragma unroll
  for (int m = 16; m >= 1; m >>= 1) {
    s += __shfl_xor(s, m, 32);
    s2 += __shfl_xor(s2, m, 32);
  }
  const float mean = s * (1.0f / HID);
  const float var = s2 * (1.0f / HID) - mean * mean;
  const float rs = rsqrtf(var + EPSV);
  const int c = lane * 4;
  float4 o;
  o.x = (v.x - mean) * rs * g[c + 0] + b[c + 0];
  o.y = (v.y - mean) * rs * g[c + 1] + b[c + 1];
  o.z = (v.z - mean) * rs * g[c + 2] + b[c + 2];
  o.w = (v.w - mean) * rs * g[c + 3] + b[c + 3];
  if (relu) {
    o.x = fmaxf(o.x, 0.0f); o.y = fmaxf(o.y, 0.0f);
    o.z = fmaxf(o.z, 0.0f); o.w = fmaxf(o.w, 0.0f);
  }
  ((float4*)(y + row * HID))[lane] = o;
}

__global__ void add_opt_relu_k(const float* __restrict__ a,
                               const float* __restrict__ b,
                               float* __restrict__ y, long n, int relu) {
  for (long i = blockIdx.x * (long)blockDim.x + threadIdx.x; i < n;
       i += (long)gridDim.x * blockDim.x) {
    float v = a[i] + b[i];
    y[i] = relu ? fmaxf(v, 0.0f) : v;
  }
}

// ------------------------- WMMA GEMM  C = A@B (+bias) ----------------------
// A: [M,128] row-major, B: [128,128] row-major, C: [M,128].
// Wave computes one 16x16 tile; 8 waves/block cover all 128 output cols.
// FP32 path: V_WMMA_F32_16X16X4_F32, K unrolled 128/4 = 32 wmma ops.
// A 16x4 lane layout (ISA 7.12.2): lane l<16 -> M=l, {K,K+1}; lane>=16 -> {K+2,K+3}.
// B 4x16: lane l<16 -> N=l rows {K,K+1}; lanes 16-31 rows {K+2,K+3}.
// C/D: vgpr r -> row r (lanes 0-15) / r+8 (lanes 16-31), col = lane&15.
__global__ __launch_bounds__(256) void gemm128_wmma_k(
    const float* __restrict__ A, const float* __restrict__ B,
    const float* __restrict__ bias, float* __restrict__ C, int M, int relu) {
  const int lane = threadIdx.x & 31;
  const int wave = threadIdx.x >> 5;  // 0..7 -> column tile
  const int row0 = blockIdx.x * 16;
  if (row0 >= M) return;
  const int col0 = wave * 16;
  const int l = lane & 15;
  const int lh = lane >> 4;  // 0 or 1

  v8f acc;
  const float binit = bias ? bias[col0 + l] : 0.0f;
#pragma unroll
  for (int r = 0; r < 8; ++r) acc[r] = binit;

  const float* __restrict__ arow = A + (size_t)(row0 + l) * HID;
#pragma unroll
  for (int k0 = 0; k0 < HID; k0 += 4) {
    const int ka = k0 + 2 * lh;
    v2f a, b;
    a.x = arow[ka];
    a.y = arow[ka + 1];
    b.x = B[(size_t)ka * HID + col0 + l];
    b.y = B[(size_t)(ka + 1) * HID + col0 + l];
    acc = __builtin_amdgcn_wmma_f32_16x16x4_f32(false, a, false, b, (short)0,
                                                acc, false, false);
  }
#pragma unroll
  for (int r = 0; r < 8; ++r) {
    const int row = row0 + r + 8 * lh;
    float v = acc[r];
    if (relu) v = fmaxf(v, 0.0f);
    C[(size_t)row * HID + col0 + l] = v;
  }
}

// --------------------------- GAT attention pieces --------------------------

// per (node, head): a_s = <h[n,h,:], a_src[h,:]>, a_d likewise
__global__ void head_dots_k(const float* __restrict__ hw,
                            const float* __restrict__ asrc,
                            const float* __restrict__ adst,
                            float* __restrict__ as_, float* __restrict__ ad_,
                            long nh) {
  for (long i = blockIdx.x * (long)blockDim.x + threadIdx.x; i < nh;
       i += (long)gridDim.x * blockDim.x) {
    const long n = i >> 2;
    const int h = (int)(i & 3);
    const float* p = hw + n * HID + h * CH;
    const float* sa = asrc + h * CH;
    const float* sd = adst + h * CH;
    float ss = 0.0f, sdd = 0.0f;
#pragma unroll
    for (int c = 0; c < CH; ++c) {
      float v = p[c];
      ss += v * sa[c];
      sdd += v * sd[c];
    }
    as_[i] = ss;
    ad_[i] = sdd;
  }
}

// pass 1: segment_max over incoming edges via ordered-uint atomicMax
__global__ void edge_max_k(const int* __restrict__ src,
                           const int* __restrict__ dst, int E, int N,
                           const float* __restrict__ as_,
                           const float* __restrict__ ad_,
                           unsigned* __restrict__ emax_u) {
  const long tot = (long)E + N;
  for (long e = blockIdx.x * (long)blockDim.x + threadIdx.x; e < tot;
       e += (long)gridDim.x * blockDim.x) {
    int s, d;
    if (e < E) { s = src[e]; d = dst[e]; } else { s = d = (int)(e - E); }
#pragma unroll
    for (int h = 0; h < NHEAD; ++h) {
      float ev = leaky02(as_[(size_t)s * NHEAD + h] + ad_[(size_t)d * NHEAD + h]);
      atomicMax(&emax_u[(size_t)d * NHEAD + h], fkey(ev));
    }
  }
}

__global__ void emax_finalize_k(const unsigned* __restrict__ emax_u,
                                float* __restrict__ emax_f, long nh) {
  for (long i = blockIdx.x * (long)blockDim.x + threadIdx.x; i < nh;
       i += (long)gridDim.x * blockDim.x) {
    unsigned u = emax_u[i];
    emax_f[i] = (u == 0u) ? 0.0f : funkey(u);  // untouched -> -inf -> 0 (ref)
  }
}

// pass 2: ex = exp(e - emax[dst]); denom[dst] += ex
__global__ void edge_exp_k(const int* __restrict__ src,
                           const int* __restrict__ dst, int E, int N,
                           const float* __restrict__ as_,
                           const float* __restrict__ ad_,
                           const float* __restrict__ emax_f,
                           float* __restrict__ ex, float* __restrict__ denom) {
  const long tot = (long)E + N;
  for (long e = blockIdx.x * (long)blockDim.x + threadIdx.x; e < tot;
       e += (long)gridDim.x * blockDim.x) {
    int s, d;
    if (e < E) { s = src[e]; d = dst[e]; } else { s = d = (int)(e - E); }
#pragma unroll
    for (int h = 0; h < NHEAD; ++h) {
      float ev = leaky02(as_[(size_t)s * NHEAD + h] + ad_[(size_t)d * NHEAD + h]);
      float x = __expf(ev - emax_f[(size_t)d * NHEAD + h]);
      ex[e * NHEAD + h] = x;
      atomicAdd(&denom[(size_t)d * NHEAD + h], x);
    }
  }
}

// pass 3: out[dst] += h[src] * alpha ; one wave per edge, lane l handles
// channels {l, l+32, l+64, l+96} -> atomics coalesced across the wave
__global__ __launch_bounds__(256) void edge_scatter_k(
    const int* __restrict__ src, const int* __restrict__ dst, int E, int N,
    const float* __restrict__ hw, const float* __restrict__ ex,
    const float* __restrict__ denom, float* __restrict__ out) {
  const int lane = threadIdx.x & 31;
  const int wpb = blockDim.x >> 5;
  const long e = (long)blockIdx.x * wpb + (threadIdx.x >> 5);
  const long tot = (long)E + N;
  if (e >= tot) return;
  int s, d;
  if (e < E) { s = src[e]; d = dst[e]; } else { s = d = (int)(e - E); }
  float alpha[NHEAD];
#pragma unroll
  for (int h = 0; h < NHEAD; ++h)
    alpha[h] = ex[e * NHEAD + h] / (denom[(size_t)d * NHEAD + h] + 1e-16f);
  const float* hs = hw + (size_t)s * HID;
  float* od = out + (size_t)d * HID;
#pragma unroll
  for (int k = 0; k < NHEAD; ++k) {
    const int ch = lane + 32 * k;  // head = k
    atomicAdd(&od[ch], hs[ch] * alpha[k]);
  }
}

// ------------------------------ pooling / head -----------------------------

__global__ __launch_bounds__(128) void pool_add_k(const float* __restrict__ h,
                                                  const int* __restrict__ batch,
                                                  float* __restrict__ g,
                                                  int nrows) {
  const int n = blockIdx.x;
  if (n >= nrows) return;
  const int c = threadIdx.x;
  atomicAdd(&g[(size_t)batch[n] * HID + c], h[(size_t)n * HID + c]);
}

__global__ void class_head_k(const float* __restrict__ g,
                             const float* __restrict__ W,  // [128,2] row-major
                             const float* __restrict__ b,
                             float* __restrict__ out, int G) {
  const int gi = blockIdx.x * blockDim.x + threadIdx.x;
  if (gi >= G) return;
  const float* row = g + (size_t)gi * HID;
  float l0 = b[0], l1 = b[1];
#pragma unroll 4
  for (int c = 0; c < HID; ++c) {
    float v = row[c];
    l0 += v * W[c * 2 + 0];
    l1 += v * W[c * 2 + 1];
  }
  const float m = fmaxf(l0, l1);
  const float lse = m + logf(expf(l0 - m) + expf(l1 - m));
  out[gi * 2 + 0] = l0 - lse;
  out[gi * 2 + 1] = l1 - lse;
}

// ------------------------------ host helpers -------------------------------

static inline int cdiv(long a, long b) { return (int)((a + b - 1) / b); }

static void run_zero(float* p, long n, hipStream_t st) {
  int grid = cdiv(n, 256); if (grid > 4096) grid = 4096;
  zero_f32_k<<<grid, 256, 0, st>>>(p, n);
}

static void run_bn(const float* x, const float* g, const float* b, float* y,
                   int nrows, float* stats, float* ss, hipStream_t st) {
  run_zero(stats, 2 * HID, st);
  int grid = nrows < 512 ? nrows : 512;
  bn_partial_k<<<grid, HID, 0, st>>>(x, nrows, stats);
  bn_finalize_k<<<1, HID, 0, st>>>(stats, g, b, 1.0f / (float)nrows, ss);
  long total = (long)nrows * HID;
  int agrid = cdiv(total, 256); if (agrid > 8192) agrid = 8192;
  bn_apply_k<<<agrid, 256, 0, st>>>(x, ss, y, total);
}

static void run_gat(const float* h, const float* W, const float* asrc,
                    const float* adst, float* hw, float* as_, float* ad_,
                    unsigned* emax_u, float* emax_f, float* denom, float* ex,
                    float* outr, const int* src, const int* dst, int E, int N,
                    hipStream_t st) {
  gemm128_wmma_k<<<N / 16, 256, 0, st>>>(h, W, nullptr, hw, N, 0);
  const long nh = (long)N * NHEAD;
  head_dots_k<<<cdiv(nh, 256), 256, 0, st>>>(hw, asrc, adst, as_, ad_, nh);
  run_zero((float*)emax_u, nh, st);  // key 0 == -inf sentinel
  run_zero(denom, nh, st);
  run_zero(outr, (long)N * HID, st);
  const long tot = (long)E + N;
  int eg = cdiv(tot, 256); if (eg > 8192) eg = 8192;
  edge_max_k<<<eg, 256, 0, st>>>(src, dst, E, N, as_, ad_, emax_u);
  emax_finalize_k<<<cdiv(nh, 256), 256, 0, st>>>(emax_u, emax_f, nh);
  edge_exp_k<<<eg, 256, 0, st>>>(src, dst, E, N, as_, ad_, emax_f, ex, denom);
  edge_scatter_k<<<cdiv(tot, 8), 256, 0, st>>>(src, dst, E, N, hw, ex, denom,
                                               outr);
}

// ------------------------------ kernel_launch ------------------------------
// Param flattening (setup_inputs insertion order):
//  0 x | 1 edge_index | 2 batch
//  3 bn_feat.g  4 bn_feat.b
//  5 conv_feat.W  6 .a_src  7 .a_dst  8 .b
//  9 bn_conv_feat.g 10 .b | 11 ln_conv_feat.g 12 .b
//  layers li=0..2 at 13+12*li: gat.W,a_src,a_dst,b, bn.g,b, ln.g,b, res.W,b,
//                              bn_res.g,b
//  49 bn_fc.g 50 .b | 51 ln_fc.g 52 .b | 53 lins.W 54 .b
//  55 bn_hidden.g 56 .b | 57 ln_hidden.g 58 .b | 59 lin_class.W 60 .b
// Note: GAT/res biases are dropped where a BatchNorm immediately follows
// (per-column constant shifts are annihilated by BN — exactly equivalent).

extern "C" void kernel_launch(void* const* d_in, const int* in_sizes, int n_in,
                              void* d_out, int out_size, void* d_ws,
                              size_t ws_size, hipStream_t stream) {
  const int N = in_sizes[0] / HID;  // 50000
  const int E = in_sizes[1] / 2;    // 800000
  const int G = 256;                // NUM_GRAPHS

  const float* x = (const float*)d_in[0];
  const int* ei = (const int*)d_in[1];
  const int* srcv = ei;
  const int* dstv = ei + E;
  const int* batch = (const int*)d_in[2];
  auto P = [&](int i) { return (const float*)d_in[i]; };

  // workspace carve-out (256B aligned slots)
  size_t off = 0;
  auto take = [&](size_t elems) {
    void* p = (char*)d_ws + off;
    off += (elems * sizeof(float) + 255) & ~(size_t)255;
    return p;
  };
  float* hA   = (float*)take((size_t)N * HID);  // current activations
  float* hw   = (float*)take((size_t)N * HID);  // x@W / residual scratch
  float* outr = (float*)take((size_t)N * HID);  // GAT aggregate
  float* as_  = (float*)take((size_t)N * NHEAD);
  float* ad_  = (float*)take((size_t)N * NHEAD);
  unsigned* emax_u = (unsigned*)take((size_t)N * NHEAD);
  float* emax_f = (float*)take((size_t)N * NHEAD);
  float* denom  = (float*)take((size_t)N * NHEAD);
  float* ex     = (float*)take(((size_t)E + N) * NHEAD);
  float* stats  = (float*)take(2 * HID);
  float* ss     = (float*)take(2 * HID);
  float* g0     = (float*)take((size_t)G * HID);
  float* g1     = (float*)take((size_t)G * HID);
  float* g2     = (float*)take((size_t)G * HID);

  const long nelem = (long)N * HID;
  const int egrid = cdiv(nelem, 256) > 8192 ? 8192 : cdiv(nelem, 256);

  // ---- stem: bn_feat -> GAT(F_IN->HID) -> bn -> ln+relu ----
  run_bn(x, P(3), P(4), hA, N, stats, ss, stream);
  run_gat(hA, P(5), P(6), P(7), hw, as_, ad_, emax_u, emax_f, denom, ex, outr,
          srcv, dstv, E, N, stream);
  run_bn(outr, P(9), P(10), outr, N, stats, ss, stream);
  layernorm128_k<<<cdiv(N, 8), 256, 0, stream>>>(outr, P(11), P(12), hA, N, 1);

  // ---- 3 residual GAT layers ----
  for (int li = 0; li < 3; ++li) {
    const int base = 13 + 12 * li;
    run_gat(hA, P(base + 0), P(base + 1), P(base + 2), hw, as_, ad_, emax_u,
            emax_f, denom, ex, outr, srcv, dstv, E, N, stream);
    run_bn(outr, P(base + 4), P(base + 5), outr, N, stats, ss, stream);
    layernorm128_k<<<cdiv(N, 8), 256, 0, stream>>>(outr, P(base + 6),
                                                   P(base + 7), outr, N, 0);
    // res = identity @ W_res (bias dropped: BN-invariant), then bn_res
    gemm128_wmma_k<<<N / 16, 256, 0, stream>>>(hA, P(base + 8), nullptr, hw, N,
                                               0);
    run_bn(hw, P(base + 10), P(base + 11), hw, N, stats, ss, stream);
    add_opt_relu_k<<<egrid, 256, 0, stream>>>(outr, hw, hA, nelem, 1);
  }

  // ---- global_add_pool ----
  run_zero(g0, (long)G * HID, stream);
  pool_add_k<<<N, HID, 0, stream>>>(hA, batch, g0, N);

  // ---- FC residual block ----
  run_bn(g0, P(49), P(50), g1, G, stats, ss, stream);
  layernorm128_k<<<cdiv(G, 8), 256, 0, stream>>>(g1, P(51), P(52), g1, G, 0);
  gemm128_wmma_k<<<G / 16, 256, 0, stream>>>(g1, P(53), P(54), g2, G, 1);
  add_opt_relu_k<<<cdiv((long)G * HID, 256), 256, 0, stream>>>(
      g2, g0, g1, (long)G * HID, 0);

  // ---- head: bn -> ln -> linear(2) -> log_softmax ----
  run_bn(g1, P(55), P(56), g1, G, stats, ss, stream);
  layernorm128_k<<<cdiv(G, 8), 256, 0, stream>>>(g1, P(57), P(58), g1, G, 0);
  class_head_k<<<1, G, 0, stream>>>(g1, P(59), P(60), (float*)d_out, G);
}